// SparseNonzeroAvgPooling_27762668601799
// MI455X (gfx1250) — compile-verified
//
#include <hip/hip_runtime.h>
#include <hip/hip_bf16.h>

#define C_CH 32

// ---------------- CDNA5 async global->LDS path (guarded) ----------------
#if __has_builtin(__builtin_amdgcn_global_load_async_to_lds_b32)
#define HAVE_ASYNC_LDS 1
#else
#define HAVE_ASYNC_LDS 0
#endif

__device__ __forceinline__ void wait_async0() {
#if __has_builtin(__builtin_amdgcn_s_wait_asynccnt)
  __builtin_amdgcn_s_wait_asynccnt(0);
#else
  asm volatile("s_wait_asynccnt 0" ::: "memory");
#endif
}

// ---------------- hot loop: one wave32 per output voxel ----------------
// Lane layout: g = lane>>3 selects one of 4 rows processed per step,
// c4 = lane&7 selects a float4 channel quad. One global_load_b128 per wave
// instruction covers 4 gathered rows (512B). Pair list staged via the CDNA5
// async global->LDS copy (ASYNCcnt-tracked), then read back as broadcasts.
__global__ __launch_bounds__(256) void reduce_avg_kernel(const float* __restrict__ feats,
                                                         const int* __restrict__ pairs,
                                                         const int* __restrict__ counts,
                                                         const int* __restrict__ ends,
                                                         float* __restrict__ out, int n_out) {
  __shared__ int stage[8][32]; // per-wave staging of the row list
  const int wid = threadIdx.x >> 5;
  const int lane = threadIdx.x & 31;
  const int o = blockIdx.x * 8 + wid; // wave-uniform
  if (o >= n_out) return;

  const int cnt = counts[o];
  const int end = ends[o]; // post-scatter cursor == exclusive offset + count
  const int start = end - cnt;
  const int g = lane >> 3; // row group 0..3
  const int c4 = lane & 7; // channel quad 0..7

  float4 acc = make_float4(0.f, 0.f, 0.f, 0.f);
  for (int base = start; base < end; base += 32) {
    const int nchunk = min(32, end - base);
    if (lane < nchunk) {
#if HAVE_ASYNC_LDS
      int* gp = const_cast<int*>(pairs + base + lane); // builtin wants non-const AS(1) ptr
      __builtin_amdgcn_global_load_async_to_lds_b32(
          (__attribute__((address_space(1))) int*)gp,
          (__attribute__((address_space(3))) int*)&stage[wid][lane],
          0, 0);
#else
      stage[wid][lane] = pairs[base + lane];
#endif
    }
#if HAVE_ASYNC_LDS
    wait_async0(); // ASYNCcnt==0 -> staged rows visible in LDS
#endif
    for (int j = 0; j < nchunk; j += 4) { // 4 rows per step, b128 per lane
      const int jg = j + g;
      if (jg < nchunk) {
        const int r = stage[wid][jg];
        const float4 v = *(const float4*)(feats + (size_t)r * C_CH + c4 * 4);
        acc.x += v.x;
        acc.y += v.y;
        acc.z += v.z;
        acc.w += v.w;
      }
    }
  }
  // Fold the 4 row-groups (lane bits 3..4) with xor-shuffles.
  for (int m = 8; m <= 16; m <<= 1) {
    acc.x += __shfl_xor(acc.x, m, 32);
    acc.y += __shfl_xor(acc.y, m, 32);
    acc.z += __shfl_xor(acc.z, m, 32);
    acc.w += __shfl_xor(acc.w, m, 32);
  }
  const float d = fmaxf((float)cnt, 1.0f);
  if (lane < 8) { // 8 lanes x float4 = one 128B row store
    float4 r;
    r.x = acc.x / d;
    r.y = acc.y / d;
    r.z = acc.z / d;
    r.w = acc.w / d;
    *(float4*)(out + (size_t)o * C_CH + lane * 4) = r;
  }
}

// ---------------- generic helpers ----------------
__global__ __launch_bounds__(256) void zero_i32_kernel(int* __restrict__ p, long long n) {
  long long i = (long long)blockIdx.x * blockDim.x + threadIdx.x;
  if (i < n) p[i] = 0;
}

// ---------------- CSR build: count / scan / scatter ----------------
__global__ __launch_bounds__(256) void count_kernel(const int* __restrict__ out_map,
                                                    int* __restrict__ counts, int M) {
  int i = blockIdx.x * blockDim.x + threadIdx.x;
  if (i < M) atomicAdd(&counts[out_map[i]], 1);
}

__global__ __launch_bounds__(256) void partial_sums_kernel(const int* __restrict__ counts,
                                                           int* __restrict__ bsums, int n) {
  __shared__ int lds[256];
  int i = blockIdx.x * 256 + threadIdx.x;
  lds[threadIdx.x] = (i < n) ? counts[i] : 0;
  __syncthreads();
  for (int off = 128; off > 0; off >>= 1) {
    if (threadIdx.x < off) lds[threadIdx.x] += lds[threadIdx.x + off];
    __syncthreads();
  }
  if (threadIdx.x == 0) bsums[blockIdx.x] = lds[0];
}

// Single-workgroup in-place exclusive scan (handles arbitrary n with a carry).
__global__ __launch_bounds__(1024) void scan_sums_kernel(int* __restrict__ data, int n) {
  __shared__ int lds[1024];
  __shared__ int carry_s;
  if (threadIdx.x == 0) carry_s = 0;
  __syncthreads();
  for (int base = 0; base < n; base += 1024) {
    int i = base + (int)threadIdx.x;
    int v = (i < n) ? data[i] : 0;
    lds[threadIdx.x] = v;
    __syncthreads();
    for (int off = 1; off < 1024; off <<= 1) {
      int t = (threadIdx.x >= (unsigned)off) ? lds[threadIdx.x - off] : 0;
      __syncthreads();
      lds[threadIdx.x] += t;
      __syncthreads();
    }
    int incl = lds[threadIdx.x];
    int carry = carry_s;
    __syncthreads();                       // all reads of carry_s done
    if (i < n) data[i] = carry + incl - v; // exclusive
    if (threadIdx.x == 1023) carry_s = carry + incl; // lds[1023] = chunk total
    __syncthreads();
  }
}

__global__ __launch_bounds__(256) void block_scan_kernel(const int* __restrict__ counts,
                                                         const int* __restrict__ boffs,
                                                         int* __restrict__ cursors, int n) {
  __shared__ int lds[256];
  int i = blockIdx.x * 256 + threadIdx.x;
  int v = (i < n) ? counts[i] : 0;
  lds[threadIdx.x] = v;
  __syncthreads();
  for (int off = 1; off < 256; off <<= 1) {
    int t = (threadIdx.x >= (unsigned)off) ? lds[threadIdx.x - off] : 0;
    __syncthreads();
    lds[threadIdx.x] += t;
    __syncthreads();
  }
  if (i < n) cursors[i] = boffs[blockIdx.x] + lds[threadIdx.x] - v; // exclusive offset
}

// Store in_map[m] (the feature row) directly: removes one indirection from reduce.
__global__ __launch_bounds__(256) void scatter_kernel(const int* __restrict__ out_map,
                                                      const int* __restrict__ in_map,
                                                      int* __restrict__ cursors,
                                                      int* __restrict__ pairs, int M) {
  int i = blockIdx.x * blockDim.x + threadIdx.x;
  if (i < M) {
    int o = out_map[i];
    int pos = atomicAdd(&cursors[o], 1); // cursors[o] ends at segment end offset
    pairs[pos] = in_map[i];
  }
}

// ---------------- fallback path (small workspace): direct atomics ----------------
__global__ __launch_bounds__(256) void direct_accum_kernel(const float* __restrict__ feats,
                                                           const int* __restrict__ in_map,
                                                           const int* __restrict__ out_map,
                                                           float* __restrict__ out,
                                                           int* __restrict__ counts, int M) {
  long long t = (long long)blockIdx.x * blockDim.x + threadIdx.x;
  int pair = (int)(t >> 5);
  int lane = (int)(t & 31);
  if (pair >= M) return;
  int o = out_map[pair];
  int r = in_map[pair];
  atomicAdd(&out[(size_t)o * C_CH + lane], feats[(size_t)r * C_CH + lane]);
  if (lane == 0) atomicAdd(&counts[o], 1);
}

__global__ __launch_bounds__(256) void divide_kernel(float* __restrict__ out,
                                                     const int* __restrict__ counts, int n_out) {
  long long t = (long long)blockIdx.x * blockDim.x + threadIdx.x;
  int o = (int)(t >> 5);
  int lane = (int)(t & 31);
  if (o >= n_out) return;
  out[(size_t)o * C_CH + lane] /= fmaxf((float)counts[o], 1.0f);
}

// ---------------- launch ----------------
extern "C" void kernel_launch(void* const* d_in, const int* in_sizes, int n_in,
                              void* d_out, int out_size, void* d_ws, size_t ws_size,
                              hipStream_t stream) {
  const float* feats = (const float*)d_in[0];
  const int* in_map = (const int*)d_in[1];
  const int* out_map = (const int*)d_in[2];
  const int M = in_sizes[1];
  const int n_out = out_size / C_CH;
  float* out = (float*)d_out;
  if (n_out <= 0 || M <= 0) return;

  // Workspace layout (all int32): counts | cursors | block_sums(padded) | pairs
  int* counts = (int*)d_ws;
  int* cursors = counts + n_out;
  const int nblk = (n_out + 255) / 256;
  int* boffs = cursors + n_out;
  int* pairs = boffs + ((nblk + 63) & ~63);
  const size_t need = ((size_t)(pairs - counts) + (size_t)M) * sizeof(int) + 16 /*slack*/;

  const int tb = 256;
  const int gM = (M + tb - 1) / tb;
  const int gN = (n_out + tb - 1) / tb;

  if (ws_size >= need) {
    zero_i32_kernel<<<gN, tb, 0, stream>>>(counts, (long long)n_out);
    count_kernel<<<gM, tb, 0, stream>>>(out_map, counts, M);
    partial_sums_kernel<<<nblk, tb, 0, stream>>>(counts, boffs, n_out);
    scan_sums_kernel<<<1, 1024, 0, stream>>>(boffs, nblk);
    block_scan_kernel<<<nblk, tb, 0, stream>>>(counts, boffs, cursors, n_out);
    scatter_kernel<<<gM, tb, 0, stream>>>(out_map, in_map, cursors, pairs, M);
    reduce_avg_kernel<<<(n_out + 7) / 8, tb, 0, stream>>>(feats, pairs, counts, cursors, out,
                                                          n_out);
  } else {
    // Small-workspace fallback: direct atomic scatter-add (needs only counts).
    long long total = (long long)n_out * C_CH;
    zero_i32_kernel<<<gN, tb, 0, stream>>>(counts, (long long)n_out);
    zero_i32_kernel<<<(int)((total + tb - 1) / tb), tb, 0, stream>>>((int*)out, total);
    long long tthreads = (long long)M * 32;
    direct_accum_kernel<<<(int)((tthreads + tb - 1) / tb), tb, 0, stream>>>(feats, in_map,
                                                                            out_map, out,
                                                                            counts, M);
    divide_kernel<<<(int)((total + tb - 1) / tb), tb, 0, stream>>>(out, counts, n_out);
  }
}